// SGC_10316511445628
// MI455X (gfx1250) — compile-verified
//
#include <hip/hip_runtime.h>

typedef float v2f __attribute__((ext_vector_type(2)));
typedef float v8f __attribute__((ext_vector_type(8)));

#define FDIM 64
#define CDIM 64
#define EPW  256   // edges per wave in the segmented SpMM

// ---------------- kernel 0: zero the h workspace ----------------
__global__ void SGC_zero_f4(float4* __restrict__ p, long n4) {
  long i = blockIdx.x * (long)blockDim.x + threadIdx.x;
  long stride = (long)gridDim.x * blockDim.x;
  float4 z = make_float4(0.f, 0.f, 0.f, 0.f);
  for (; i < n4; i += stride) p[i] = z;
}

// ---------------- kernel 1: segmented SpMM (rows sorted) ----------------
// One wave handles EPW consecutive edges. Each lane owns features [2*lane, 2*lane+1].
// Rows fully interior to the chunk get plain stores; only the chunk's first/last
// row (which may straddle chunk boundaries) use f32 atomics.
__global__ void SGC_spmm_seg(const float* __restrict__ feat,
                             const int*   __restrict__ erow,
                             const int*   __restrict__ ecol,
                             const float* __restrict__ eval,
                             float* __restrict__ h,
                             int nEdges, int nWaves) {
  int gwave = (int)((blockIdx.x * (long)blockDim.x + threadIdx.x) >> 5);
  if (gwave >= nWaves) return;                 // wave-uniform guard
  int lane = threadIdx.x & 31;

  long base = (long)gwave * EPW;
  int cnt = nEdges - (int)base;
  if (cnt > EPW) cnt = EPW;

  const int firstRow = erow[base];
  const int lastRow  = erow[base + cnt - 1];

  int   cur = firstRow;
  float ax = 0.f, ay = 0.f;

  for (int i = 0; i < cnt; ++i) {
    long e = base + i;
    int  r = erow[e];                          // wave-uniform
    if (r != cur) {
      float2* p = (float2*)(h + (long)cur * FDIM) + lane;
      if (cur == firstRow || cur == lastRow) { // may straddle chunks -> atomic
        atomicAdd(&p->x, ax);
        atomicAdd(&p->y, ay);
      } else {                                 // fully owned -> plain store
        *p = make_float2(ax, ay);
      }
      ax = 0.f; ay = 0.f; cur = r;
    }
    float v = eval[e];                         // wave-uniform
    int   c = ecol[e];                         // wave-uniform
    float2 f = ((const float2*)(feat + (long)c * FDIM))[lane];
    ax = fmaf(v, f.x, ax);
    ay = fmaf(v, f.y, ay);
  }
  // final flush: cur == lastRow here, always potentially shared -> atomic
  float2* p = (float2*)(h + (long)cur * FDIM) + lane;
  atomicAdd(&p->x, ax);
  atomicAdd(&p->y, ay);
}

// ---------------- kernel 2: out = h @ W^T + b via f32 WMMA ----------------
// One wave per 16-row tile; 4 v8f accumulators cover all 64 classes.
// A frag (16x4 f32, ISA 7.12.2): lanes 0-15 hold M=lane, K={k0,k0+1};
//                                lanes 16-31 hold M=lane-16, K={k0+2,k0+3}.
// B frag (4x16 f32): VGPR0 = row K (lanes0-15: k0 | lanes16-31: k0+2), N=lane%16;
//                    VGPR1 = next K. Same index formula as A by symmetry.
// C/D (16x16 f32): VGPR v -> M = v + 8*(lane>=16), N = lane%16.
__global__ void SGC_gemm_wmma(const float* __restrict__ h,
                              const float* __restrict__ W,
                              const float* __restrict__ bias,
                              float* __restrict__ out,
                              int nRows, int nTiles) {
  int wave = (int)((blockIdx.x * (long)blockDim.x + threadIdx.x) >> 5);
  if (wave >= nTiles) return;                  // wave-uniform: EXEC stays all-1s
  int lane = threadIdx.x & 31;
  int l16  = lane & 15;
  int lh   = lane >> 4;
  int m0   = wave * 16;

  v8f c0, c1, c2, c3;
  {
    float b0 = bias[ 0 + l16], b1 = bias[16 + l16];
    float b2 = bias[32 + l16], b3 = bias[48 + l16];
#pragma unroll
    for (int v = 0; v < 8; ++v) { c0[v] = b0; c1[v] = b1; c2[v] = b2; c3[v] = b3; }
  }

  int arow = m0 + l16;
  if (arow >= nRows) arow = nRows - 1;         // clamp (stores are guarded below)
  const float* hrow = h + (long)arow * FDIM;

#pragma unroll
  for (int k0 = 0; k0 < FDIM; k0 += 4) {
    int kk = k0 + 2 * lh;
    v2f a;  a.x  = hrow[kk];                 a.y  = hrow[kk + 1];
    v2f b0v; b0v.x = W[( 0 + l16) * FDIM + kk]; b0v.y = W[( 0 + l16) * FDIM + kk + 1];
    v2f b1v; b1v.x = W[(16 + l16) * FDIM + kk]; b1v.y = W[(16 + l16) * FDIM + kk + 1];
    v2f b2v; b2v.x = W[(32 + l16) * FDIM + kk]; b2v.y = W[(32 + l16) * FDIM + kk + 1];
    v2f b3v; b3v.x = W[(48 + l16) * FDIM + kk]; b3v.y = W[(48 + l16) * FDIM + kk + 1];
    c0 = __builtin_amdgcn_wmma_f32_16x16x4_f32(false, a, false, b0v, (short)0, c0, false, false);
    c1 = __builtin_amdgcn_wmma_f32_16x16x4_f32(false, a, false, b1v, (short)0, c1, false, false);
    c2 = __builtin_amdgcn_wmma_f32_16x16x4_f32(false, a, false, b2v, (short)0, c2, false, false);
    c3 = __builtin_amdgcn_wmma_f32_16x16x4_f32(false, a, false, b3v, (short)0, c3, false, false);
  }

#pragma unroll
  for (int v = 0; v < 8; ++v) {
    int m = m0 + v + 8 * lh;
    if (m < nRows) {
      float* o = out + (long)m * CDIM;
      o[ 0 + l16] = c0[v];
      o[16 + l16] = c1[v];
      o[32 + l16] = c2[v];
      o[48 + l16] = c3[v];
    }
  }
}

extern "C" void kernel_launch(void* const* d_in, const int* in_sizes, int n_in,
                              void* d_out, int out_size, void* d_ws, size_t ws_size,
                              hipStream_t stream) {
  const float* feat = (const float*)d_in[0];
  const int*   erow = (const int*)  d_in[1];
  const int*   ecol = (const int*)  d_in[2];
  const float* eval = (const float*)d_in[3];
  const float* W    = (const float*)d_in[4];
  const float* bias = (const float*)d_in[5];

  const int N = in_sizes[0] / FDIM;
  const int E = in_sizes[1];

  float* h   = (float*)d_ws;      // N*FDIM floats of scratch
  float* out = (float*)d_out;

  // 1) zero h
  long n4 = (long)N * FDIM / 4;
  SGC_zero_f4<<<1024, 256, 0, stream>>>((float4*)h, n4);

  // 2) segmented SpMM
  int nWaves = (E + EPW - 1) / EPW;
  int sBlocks = (nWaves * 32 + 255) / 256;
  SGC_spmm_seg<<<sBlocks, 256, 0, stream>>>(feat, erow, ecol, eval, h, E, nWaves);

  // 3) WMMA GEMM + bias
  int nTiles = (N + 15) / 16;
  int gBlocks = (nTiles * 32 + 255) / 256;
  SGC_gemm_wmma<<<gBlocks, 256, 0, stream>>>(h, W, bias, out, N, nTiles);
}